// Criterion_24773371363774
// MI455X (gfx1250) — compile-verified
//
#include <hip/hip_runtime.h>
#include <hip/hip_bf16.h>

typedef __attribute__((ext_vector_type(2))) float v2f;
typedef __attribute__((ext_vector_type(8))) float v8f;

#define MARGIN_F 0.2f
#define EPS_F 1e-8f
#define DIM 128

// d_ws layout (floats): [0]=total loss, [1]=pair count, [2 .. 2+rows) = row norms

__global__ void zero_acc_kernel(float* ws) {
    ws[0] = 0.0f;
    ws[1] = 0.0f;
}

// One wave (32 lanes) per row: each lane loads float4, reduces across wave.
__global__ void row_norms_kernel(const float* __restrict__ batch,
                                 float* __restrict__ norms, int rows) {
    int wave = (int)((blockIdx.x * blockDim.x + threadIdx.x) >> 5);
    int lane = threadIdx.x & 31;
    if (wave >= rows) return;
    const float4* row = (const float4*)(batch + (size_t)wave * DIM);
    float4 v = row[lane];
    float s = v.x * v.x + v.y * v.y + v.z * v.z + v.w * v.w;
    #pragma unroll
    for (int off = 16; off > 0; off >>= 1) s += __shfl_xor(s, off, 32);
    if (lane == 0) norms[wave] = s;
}

// Select element k (0..7) of a v8f accumulator via cndmask chain.
__device__ __forceinline__ float pick8(v8f v, int k) {
    float r = v[0];
    r = (k == 1) ? v[1] : r;
    r = (k == 2) ? v[2] : r;
    r = (k == 3) ? v[3] : r;
    r = (k == 4) ? v[4] : r;
    r = (k == 5) ? v[5] : r;
    r = (k == 6) ? v[6] : r;
    r = (k == 7) ? v[7] : r;
    return r;
}

// Each wave handles 16 triplets. Two f32 16x16x4 WMMA accumulators build the
// Gram blocks anchors·posT and anchors·negT over K=128; diagonals are the dot
// products. d^2 = ||a||^2 + ||x||^2 - 2 a.x  (norms precomputed).
__global__ void triplet_loss_kernel(const float* __restrict__ batch,
                                    const int* __restrict__ labels,
                                    const int* __restrict__ triplets,
                                    const float* __restrict__ beta,
                                    const float* __restrict__ norms,
                                    float* __restrict__ acc, int T) {
    __shared__ float s_tot;
    __shared__ float s_cnt;
    if (threadIdx.x == 0) { s_tot = 0.0f; s_cnt = 0.0f; }
    __syncthreads();

    const int lane = threadIdx.x & 31;
    const int waveInBlock = threadIdx.x >> 5;
    const int base = (blockIdx.x * (blockDim.x >> 5) + waveInBlock) * 16;

    if (base < T) {                          // wave-uniform branch: EXEC stays full
        const int li = lane & 15;            // which of the 16 triplets this lane serves
        int t = base + li;
        int tc = (t < T) ? t : (T - 1);      // clamp tail loads (masked later)

        const int ai = triplets[tc * 3 + 0];
        const int pi = triplets[tc * 3 + 1];
        const int ni = triplets[tc * 3 + 2];

        // A 16x4 f32 layout: lane l -> row (l&15), K-offsets {2*(l>>4), +1}.
        // B 4x16 layout is the mirror: same per-lane addressing on the B row.
        const int koff = (lane >> 4) * 2;
        const float* aptr = batch + (size_t)ai * DIM + koff;
        const float* pptr = batch + (size_t)pi * DIM + koff;
        const float* nptr = batch + (size_t)ni * DIM + koff;

        v8f accp = {};
        v8f accn = {};
        #pragma unroll 8
        for (int kb = 0; kb < DIM; kb += 4) {
            v2f av = *(const v2f*)(aptr + kb);
            v2f pv = *(const v2f*)(pptr + kb);
            v2f nv = *(const v2f*)(nptr + kb);
            accp = __builtin_amdgcn_wmma_f32_16x16x4_f32(
                false, av, false, pv, (short)0, accp, false, false);
            accn = __builtin_amdgcn_wmma_f32_16x16x4_f32(
                false, av, false, nv, (short)0, accn, false, false);
        }

        // Diagonal (i,i): i<8 -> VGPR i, lane i ; i>=8 -> VGPR i-8, lane i+16.
        // Valid lanes: {0..7} U {24..31}; their (lane&15) == their triplet i.
        const int k = lane & 7;
        float dotp = pick8(accp, k);
        float dotn = pick8(accn, k);

        const bool validLane = (lane < 8) || (lane >= 24);
        const bool validTrip = (base + li) < T;

        float contrib = 0.0f;
        float ccnt = 0.0f;
        if (validLane && validTrip) {
            float na = norms[ai];
            float np_ = norms[pi];
            float nn_ = norms[ni];
            float b = beta[labels[ai]];
            float dap = sqrtf(fmaxf(na + np_ - 2.0f * dotp, 0.0f) + EPS_F);
            float dan = sqrtf(fmaxf(na + nn_ - 2.0f * dotn, 0.0f) + EPS_F);
            float pl = fmaxf(dap - b + MARGIN_F, 0.0f);
            float nl = fmaxf(b - dan + MARGIN_F, 0.0f);
            contrib = pl + nl;
            ccnt = (pl > 0.0f ? 1.0f : 0.0f) + (nl > 0.0f ? 1.0f : 0.0f);
        }

        #pragma unroll
        for (int off = 16; off > 0; off >>= 1) {
            contrib += __shfl_xor(contrib, off, 32);
            ccnt    += __shfl_xor(ccnt, off, 32);
        }
        if (lane == 0) {
            atomicAdd(&s_tot, contrib);
            atomicAdd(&s_cnt, ccnt);
        }
    }
    __syncthreads();
    if (threadIdx.x == 0) {
        atomicAdd(&acc[0], s_tot);
        atomicAdd(&acc[1], s_cnt);
    }
}

__global__ void finalize_kernel(const float* __restrict__ acc,
                                float* __restrict__ out) {
    float total = acc[0];
    float cnt = acc[1];
    out[0] = (cnt > 0.0f) ? (total / fmaxf(cnt, 1.0f)) : total;
}

extern "C" void kernel_launch(void* const* d_in, const int* in_sizes, int n_in,
                              void* d_out, int out_size, void* d_ws, size_t ws_size,
                              hipStream_t stream) {
    const float* batch    = (const float*)d_in[0];
    const int*   labels   = (const int*)d_in[1];
    const int*   triplets = (const int*)d_in[2];
    const float* beta     = (const float*)d_in[3];

    const int rows = in_sizes[0] / DIM;     // 4096
    const int T    = in_sizes[2] / 3;       // 500000

    float* ws    = (float*)d_ws;
    float* norms = ws + 2;

    zero_acc_kernel<<<1, 1, 0, stream>>>(ws);

    // one wave per row, 8 waves per 256-thread block
    int normBlocks = (rows * 32 + 255) / 256;
    row_norms_kernel<<<normBlocks, 256, 0, stream>>>(batch, norms, rows);

    // 8 waves/block * 16 triplets/wave = 128 triplets per block
    int lossBlocks = (T + 127) / 128;
    triplet_loss_kernel<<<lossBlocks, 256, 0, stream>>>(
        batch, labels, triplets, beta, norms, ws, T);

    finalize_kernel<<<1, 1, 0, stream>>>(ws, (float*)d_out);
}